// ColorHistogramLayer_16827681866032
// MI455X (gfx1250) — compile-verified
//
#include <hip/hip_runtime.h>

// ---------------------------------------------------------------------------
// ColorHistogramLayer for MI455X (gfx1250, wave32)
//   x:     [64, 3, 512, 512] f32 in [0,1)
//   feat:  per-(b,c) 16-bin normalized histograms -> [64, 48]
//   out:   relu(feat @ fc_w.T + fc_b) -> [64, 64] f32
//
// Memory-bound: 201 MB single pass (~8.6 us at 23.3 TB/s). Histogram uses
// 128-bit loads + per-wave LDS histograms (ds_add_u32) + 16 global atomics
// per block. FC uses V_WMMA_F32_16X16X4_F32 (full f32 precision).
// ---------------------------------------------------------------------------

#define BINS 16
#define CH 3
#define FEAT_DIM (BINS * CH)     // 48
#define OUT_DIM 64
#define BATCH 64
#define PLANE_PIX (512 * 512)    // 262144 pixels per (b,c) plane
#define PLANE_F4 (PLANE_PIX / 4) // 65536 float4 per plane
#define CHUNKS_PER_PLANE 16
#define CHUNK_F4 (PLANE_F4 / CHUNKS_PER_PLANE) // 4096 float4 per block
#define ITERS (CHUNK_F4 / 256)                 // 16 float4 per thread

typedef float v2f __attribute__((ext_vector_type(2)));
typedef float v8f __attribute__((ext_vector_type(8)));

// ---------------------------------------------------------------------------
// Kernel 0: zero the count table in workspace (d_ws is poisoned to 0xAA).
// ---------------------------------------------------------------------------
__global__ void zero_counts_kernel(unsigned int* __restrict__ counts, int n) {
    int i = blockIdx.x * blockDim.x + threadIdx.x;
    if (i < n) counts[i] = 0u;
}

// ---------------------------------------------------------------------------
// Kernel 1: streaming histogram.
// grid = planes(192) * CHUNKS_PER_PLANE(16) blocks, 256 threads (8 wave32).
// Per-wave privatized LDS histograms -> ds_add_u32 with low contention,
// then 16 global_atomic_add_u32 per block.
// ---------------------------------------------------------------------------
__global__ __launch_bounds__(256) void hist_kernel(const float4* __restrict__ x4,
                                                   unsigned int* __restrict__ counts) {
    __shared__ unsigned int h[8][BINS]; // 8 waves x 16 bins
    const int tid = threadIdx.x;
    if (tid < 8 * BINS) ((unsigned int*)h)[tid] = 0u;
    __syncthreads();

    const int wave = tid >> 5;
    const unsigned int plane = blockIdx.x >> 4;  // 0..191  (b*3 + c)
    const unsigned int chunk = blockIdx.x & 15;  // 0..15

    // base index in float4 units; consecutive lanes -> consecutive float4
    // (coalesced global_load_b128)
    size_t base = (size_t)plane * PLANE_F4 + (size_t)chunk * CHUNK_F4 + (size_t)tid;

    unsigned int* __restrict__ myh = h[wave];

#pragma unroll
    for (int i = 0; i < ITERS; ++i) {
        float4 v = x4[base + (size_t)i * 256];
        // values are in [0,1): floor(v*16) via int cast (non-negative),
        // clamp to 15 to match reference clip semantics.
        int b0 = (int)(v.x * (float)BINS); b0 = b0 > (BINS - 1) ? (BINS - 1) : b0;
        int b1 = (int)(v.y * (float)BINS); b1 = b1 > (BINS - 1) ? (BINS - 1) : b1;
        int b2 = (int)(v.z * (float)BINS); b2 = b2 > (BINS - 1) ? (BINS - 1) : b2;
        int b3 = (int)(v.w * (float)BINS); b3 = b3 > (BINS - 1) ? (BINS - 1) : b3;
        atomicAdd(&myh[b0], 1u);
        atomicAdd(&myh[b1], 1u);
        atomicAdd(&myh[b2], 1u);
        atomicAdd(&myh[b3], 1u);
    }
    __syncthreads();

    if (tid < BINS) {
        unsigned int s = 0;
#pragma unroll
        for (int w = 0; w < 8; ++w) s += h[w][tid];
        const unsigned int b = plane / CH;
        const unsigned int c = plane % CH;
        atomicAdd(&counts[b * FEAT_DIM + c * BINS + tid], s);
    }
}

// ---------------------------------------------------------------------------
// Kernel 2: FC + bias + ReLU via V_WMMA_F32_16X16X4_F32.
// out[64,64] = relu(feat[64,48] @ fc_w.T[48,64] + fc_b)
// One block, 16 wave32; wave w owns 16x16 tile (mt = w/4, nt = w%4).
// 12 K-steps of K=4.
//
// Fragment layouts (ISA 7.12.2):
//   A 16x4 f32 (2 VGPR): lanes 0-15 M=lane, VGPR{0,1}=K{0,1};
//                        lanes 16-31 M=lane-16, VGPR{0,1}=K{2,3}
//   B 4x16 f32 (2 VGPR): VGPR r, lanes 0-15 -> K=r, N=lane;
//                        lanes 16-31 -> K=r+2, N=lane-16
//   C/D 16x16 f32 (8 VGPR): VGPR r, lanes 0-15 -> M=r; lanes 16-31 -> M=r+8
// ---------------------------------------------------------------------------
__global__ __launch_bounds__(512) void fc_wmma_kernel(const unsigned int* __restrict__ counts,
                                                      const float* __restrict__ fc_w,
                                                      const float* __restrict__ fc_b,
                                                      float* __restrict__ out) {
    const float inv_n = 1.0f / (float)PLANE_PIX;

    const int lane = threadIdx.x & 31;
    const int wave = threadIdx.x >> 5;   // 0..15
    const int mt = wave >> 2;            // 0..3 -> rows (batch) tile
    const int nt = wave & 3;             // 0..3 -> cols (out feature) tile
    const int laneLow = lane & 15;
    const int hiSel = lane >> 4;         // 0 for lanes 0-15, 1 for 16-31

    const int n = nt * 16 + laneLow;     // output column this lane owns in B/C/D
    const float bias = fc_b[n];

    v8f c;
#pragma unroll
    for (int i = 0; i < 8; ++i) c[i] = bias;   // bias depends only on N

    const int rowA = mt * 16 + laneLow;  // batch row this lane owns in A

#pragma unroll
    for (int kk = 0; kk < FEAT_DIM / 4; ++kk) {
        const int kbase = kk * 4;
        const int ka = kbase + hiSel * 2;

        v2f a;
        a.x = (float)counts[rowA * FEAT_DIM + ka + 0] * inv_n;
        a.y = (float)counts[rowA * FEAT_DIM + ka + 1] * inv_n;

        // B[k][n] = fc_w[n][k]  (fc_w is [OUT_DIM, FEAT_DIM] row-major)
        v2f b;
        b.x = fc_w[n * FEAT_DIM + kbase + hiSel * 2 + 0];
        b.y = fc_w[n * FEAT_DIM + kbase + hiSel * 2 + 1];

        c = __builtin_amdgcn_wmma_f32_16x16x4_f32(
            /*neg_a=*/false, a, /*neg_b=*/false, b,
            /*c_mod=*/(short)0, c, /*reuse_a=*/false, /*reuse_b=*/false);
    }

#pragma unroll
    for (int r = 0; r < 8; ++r) {
        float v = c[r] > 0.0f ? c[r] : 0.0f;             // ReLU
        const int row = mt * 16 + r + hiSel * 8;         // C/D layout
        out[row * OUT_DIM + nt * 16 + laneLow] = v;
    }
}

// ---------------------------------------------------------------------------
extern "C" void kernel_launch(void* const* d_in, const int* in_sizes, int n_in,
                              void* d_out, int out_size, void* d_ws, size_t ws_size,
                              hipStream_t stream) {
    const float* x = (const float*)d_in[0];     // [64,3,512,512]
    const float* fc_w = (const float*)d_in[1];  // [64,48]
    const float* fc_b = (const float*)d_in[2];  // [64]
    float* out = (float*)d_out;                 // [64,64]

    unsigned int* counts = (unsigned int*)d_ws; // [64,48] u32

    const int planes = in_sizes[0] / PLANE_PIX; // 192
    const int n_counts = BATCH * FEAT_DIM;      // 3072

    zero_counts_kernel<<<(n_counts + 255) / 256, 256, 0, stream>>>(counts, n_counts);

    hist_kernel<<<planes * CHUNKS_PER_PLANE, 256, 0, stream>>>(
        (const float4*)x, counts);

    fc_wmma_kernel<<<1, 512, 0, stream>>>(counts, fc_w, fc_b, out);
}